// GeoMix2_33440615367381
// MI455X (gfx1250) — compile-verified
//
#include <hip/hip_runtime.h>
#include <hip/hip_bf16.h>
#include <stdint.h>

// APPNP-style propagation on MI455X (gfx1250, wave32).
// Inputs: d_in[0]=x (N*128 f32), d_in[1]=y (N*64 f32), d_in[2]=edge_index (2*E int64)
// Output: d_out = final x (N*128) ++ final y (N*64), f32.
//
// Memory/atomic-bound sparse scatter-gather: per hop, 3.2M random 768B row
// gathers + the same volume of f32 atomic scatter-adds. CDNA5 features used:
// wave32 feature-parallel rows (float4/lane == one 128-col row per wave),
// double-buffered async global->LDS staging of edge tuples (ASYNCcnt,
// in-order completion lets s_wait_asynccnt 3 overlap next-chunk index fetch
// with current-chunk compute), global_prefetch_b8 row warming, and
// non-returning global_atomic_add_f32.

namespace {
constexpr int    kN     = 100000;
constexpr int    kE     = 3200000;
constexpr int    kDX    = 128;
constexpr int    kDY    = 64;
constexpr int    kHops  = 10;
constexpr float  kAlpha = 0.1f;
constexpr float  kBeta  = 1.0f - kAlpha;   // 0.9
}

// ---------------------------------------------------------------- degree ----
__global__ void k_init_deg(float* __restrict__ deg) {
    int i = blockIdx.x * blockDim.x + threadIdx.x;
    if (i < kN) deg[i] = 1.0f;              // self loop
}

__global__ void k_accum_deg(const long long* __restrict__ dstI,
                            float* __restrict__ deg) {
    for (int e = blockIdx.x * blockDim.x + threadIdx.x; e < kE;
         e += gridDim.x * blockDim.x) {
        atomicAdd(&deg[dstI[e]], 1.0f);
    }
}

__global__ void k_dinv(float* __restrict__ deg) {
    int i = blockIdx.x * blockDim.x + threadIdx.x;
    if (i < kN) deg[i] = rsqrtf(deg[i]);    // deg >= 1 always
}

__global__ void k_edge_w(const long long* __restrict__ srcI,
                         const long long* __restrict__ dstI,
                         const float* __restrict__ dinv,
                         float* __restrict__ w) {
    int e = blockIdx.x * blockDim.x + threadIdx.x;
    if (e < kE) w[e] = kBeta * dinv[srcI[e]] * dinv[dstI[e]];
}

// ------------------------------------------------- per-hop initialization ---
// out = alpha*orig + beta*dinv[n]^2*cur   (folds the self-loop edge; fully
// initializes the output buffer so the edge pass can atomically accumulate).
__global__ void k_hop_init(const float* __restrict__ x0,
                           const float* __restrict__ y0,
                           const float* __restrict__ curX,
                           const float* __restrict__ curY,
                           const float* __restrict__ dinv,
                           float* __restrict__ out) {
    constexpr int per = (kDX + kDY) / 4;    // 48 float4 per node
    int tid = blockIdx.x * blockDim.x + threadIdx.x;
    if (tid >= kN * per) return;
    int node = tid / per;
    int q    = tid - node * per;
    float s  = dinv[node];
    float g  = kBeta * s * s;
    if (q < kDX / 4) {
        size_t idx = (size_t)node * (kDX / 4) + q;
        float4 o = ((const float4*)x0)[idx];
        float4 c = ((const float4*)curX)[idx];
        ((float4*)out)[idx] = make_float4(fmaf(g, c.x, kAlpha * o.x),
                                          fmaf(g, c.y, kAlpha * o.y),
                                          fmaf(g, c.z, kAlpha * o.z),
                                          fmaf(g, c.w, kAlpha * o.w));
    } else {
        int qy = q - kDX / 4;
        size_t idx = (size_t)node * (kDY / 4) + qy;
        float4 o = ((const float4*)y0)[idx];
        float4 c = ((const float4*)curY)[idx];
        float4* outY4 = (float4*)(out + (size_t)kN * kDX);
        outY4[idx] = make_float4(fmaf(g, c.x, kAlpha * o.x),
                                 fmaf(g, c.y, kAlpha * o.y),
                                 fmaf(g, c.z, kAlpha * o.z),
                                 fmaf(g, c.w, kAlpha * o.w));
    }
}

// --------------------------------------------------------- edge scatter -----
// One wave (32 lanes) per edge; 256-edge chunks double-buffered in LDS via
// CDNA5 async global->LDS loads. Each thread always issues exactly 3 async
// ops per staged chunk (tail addresses clamped in-bounds), so
// s_wait_asynccnt 0x3 is a precise "previous batch complete" wait and the
// next chunk's index stream overlaps the gather/atomic pipeline.
__global__ void k_edge_scatter(const long long* __restrict__ srcI,
                               const long long* __restrict__ dstI,
                               const float* __restrict__ ew,
                               const float* __restrict__ curX,
                               const float* __restrict__ curY,
                               float* __restrict__ out,
                               int nEdges) {
    __shared__ long long sSrc[2][256];
    __shared__ long long sDst[2][256];
    __shared__ float     sW[2][256];

    float* outX = out;
    float* outY = out + (size_t)kN * kDX;

    const int tid  = threadIdx.x;       // 0..255
    const int lane = tid & 31;
    const int wv   = tid >> 5;          // 0..7 waves per block
    const int nChunks = (nEdges + 255) >> 8;

    int buf   = 0;
    int chunk = blockIdx.x;

    // Stage the first chunk for this block.
    if (chunk < nChunks) {
        int e = (chunk << 8) + tid;
        if (e >= nEdges) e = nEdges - 1;                 // clamp: always 3 ops
        unsigned laS = (unsigned)(uintptr_t)&sSrc[0][tid];
        unsigned laD = (unsigned)(uintptr_t)&sDst[0][tid];
        unsigned laW = (unsigned)(uintptr_t)&sW[0][tid];
        unsigned long long gaS = (unsigned long long)(uintptr_t)(srcI + e);
        unsigned long long gaD = (unsigned long long)(uintptr_t)(dstI + e);
        unsigned long long gaW = (unsigned long long)(uintptr_t)(ew + e);
        asm volatile("global_load_async_to_lds_b64 %0, %1, off"
                     :: "v"(laS), "v"(gaS) : "memory");
        asm volatile("global_load_async_to_lds_b64 %0, %1, off"
                     :: "v"(laD), "v"(gaD) : "memory");
        asm volatile("global_load_async_to_lds_b32 %0, %1, off"
                     :: "v"(laW), "v"(gaW) : "memory");
    }

    for (; chunk < nChunks; chunk += gridDim.x) {
        const int base    = chunk << 8;
        const int nextChk = chunk + gridDim.x;
        const bool hasNext = nextChk < nChunks;          // block-uniform

        if (hasNext) {                                   // stage chunk k+1
            int e = (nextChk << 8) + tid;
            if (e >= nEdges) e = nEdges - 1;
            int nb = buf ^ 1;
            unsigned laS = (unsigned)(uintptr_t)&sSrc[nb][tid];
            unsigned laD = (unsigned)(uintptr_t)&sDst[nb][tid];
            unsigned laW = (unsigned)(uintptr_t)&sW[nb][tid];
            unsigned long long gaS = (unsigned long long)(uintptr_t)(srcI + e);
            unsigned long long gaD = (unsigned long long)(uintptr_t)(dstI + e);
            unsigned long long gaW = (unsigned long long)(uintptr_t)(ew + e);
            asm volatile("global_load_async_to_lds_b64 %0, %1, off"
                         :: "v"(laS), "v"(gaS) : "memory");
            asm volatile("global_load_async_to_lds_b64 %0, %1, off"
                         :: "v"(laD), "v"(gaD) : "memory");
            asm volatile("global_load_async_to_lds_b32 %0, %1, off"
                         :: "v"(laW), "v"(gaW) : "memory");
            // In-order async completion: waiting to <=3 guarantees the
            // previous chunk's 3 ops (issued earlier) have landed in LDS.
            asm volatile("s_wait_asynccnt 0x3" ::: "memory");
        } else {
            asm volatile("s_wait_asynccnt 0x0" ::: "memory");
        }
        __syncthreads();

        // Warm the gather rows for this chunk while waves spin up.
        {
            long long dp = sDst[buf][tid];
            __builtin_prefetch(curX + (size_t)dp * kDX, 0, 3);
            __builtin_prefetch(curY + (size_t)dp * kDY, 0, 3);
        }

        const int limit = (nEdges - base < 256) ? (nEdges - base) : 256;
        const int hi = ((wv + 1) << 5) < limit ? ((wv + 1) << 5) : limit;
        for (int i = wv << 5; i < hi; ++i) {
            long long s = sSrc[buf][i];
            long long d = sDst[buf][i];
            float we    = sW[buf][i];

            float4 v = ((const float4*)(curX + (size_t)d * kDX))[lane];
            float* orx = outX + (size_t)s * kDX + lane * 4;
            atomicAdd(orx + 0, we * v.x);
            atomicAdd(orx + 1, we * v.y);
            atomicAdd(orx + 2, we * v.z);
            atomicAdd(orx + 3, we * v.w);

            float2 vy = ((const float2*)(curY + (size_t)d * kDY))[lane];
            float* ory = outY + (size_t)s * kDY + lane * 2;
            atomicAdd(ory + 0, we * vy.x);
            atomicAdd(ory + 1, we * vy.y);
        }
        __syncthreads();
        buf ^= 1;
    }
}

// ------------------------------------------------------------- launcher -----
extern "C" void kernel_launch(void* const* d_in, const int* in_sizes, int n_in,
                              void* d_out, int out_size, void* d_ws, size_t ws_size,
                              hipStream_t stream) {
    (void)in_sizes; (void)n_in; (void)out_size; (void)ws_size;

    const float*     x0   = (const float*)d_in[0];
    const float*     y0   = (const float*)d_in[1];
    const long long* ei   = (const long long*)d_in[2];   // jnp.int64 [2, E]
    const long long* srcI = ei;
    const long long* dstI = ei + kE;
    float*           out  = (float*)d_out;

    // Workspace layout (page aligned): dinv | edge weights | ping buffer
    char*  ws   = (char*)d_ws;
    float* dinv = (float*)ws;                         // N floats   (400 KB)
    float* w    = (float*)(ws + 401408);              // E floats   (12.8 MB)
    float* bufA = (float*)(ws + 13201408);            // N*192 f32  (76.8 MB)

    // ---- normalization ----
    k_init_deg <<<(kN + 255) / 256, 256, 0, stream>>>(dinv);
    k_accum_deg<<<2048,             256, 0, stream>>>(dstI, dinv);
    k_dinv     <<<(kN + 255) / 256, 256, 0, stream>>>(dinv);
    k_edge_w   <<<(kE + 255) / 256, 256, 0, stream>>>(srcI, dstI, dinv, w);

    // ---- 10 hops, ping-pong so hop 10 lands in d_out ----
    constexpr int initBlocks = (kN * ((kDX + kDY) / 4) + 255) / 256;
    for (int h = 0; h < kHops; ++h) {
        float* o = (h & 1) ? out : bufA;              // h=9 -> d_out
        const float *cx, *cy;
        if (h == 0) { cx = x0; cy = y0; }
        else {
            const float* c = (h & 1) ? bufA : out;    // previous hop's output
            cx = c;
            cy = c + (size_t)kN * kDX;
        }
        k_hop_init    <<<initBlocks, 256, 0, stream>>>(x0, y0, cx, cy, dinv, o);
        k_edge_scatter<<<2048,       256, 0, stream>>>(srcI, dstI, w, cx, cy, o, kE);
    }
}